// NatPnModel_29798483100382
// MI455X (gfx1250) — compile-verified
//
#include <hip/hip_runtime.h>
#include <hip/hip_bf16.h>
#include <math.h>

// Problem constants (match reference)
#define NS 8192   // samples
#define DD 64     // embedding dim
#define CC 100    // classes
#define TT 30     // flow transforms

#define LOG2PI      1.8378770664093453f
#define EV_BUDGET_F 80.99277687378737f   // 0.5*64*log(4*pi)
#define LOG_EV_CLAMP_F 10.0f

typedef __attribute__((ext_vector_type(2))) float v2f;
typedef __attribute__((ext_vector_type(4))) float v4f;
typedef __attribute__((ext_vector_type(8))) float v8f;

__device__ __forceinline__ float softplus_f(float v) {
    // jax.nn.softplus: log1p(exp(-|x|)) + max(x, 0)
    return log1pf(expf(-fabsf(v))) + fmaxf(v, 0.0f);
}

// ---------------------------------------------------------------------------
// Kernel 1: radial-flow log-prob. One wave = 32 samples of one class.
// Per-class params staged in LDS (wave-uniform -> ds_load_b128 broadcast).
// z and diff live entirely in VGPRs as <2 x float> so the backend can select
// packed V_PK_FMA_F32 / V_PK_ADD_F32 (2 f32 lanes per VOP3P instruction).
// Transcendentals use raw v_log/v_sqrt/v_rcp (TRANS pipe co-executes w/ VALU).
// grid = (NS/128, CC), block = 128 (4 waves)
// ---------------------------------------------------------------------------
__global__ __launch_bounds__(128) void flow_lp_kernel(
    const float* __restrict__ x,      // [NS, DD]
    const float* __restrict__ z0,     // [CC, TT, DD]
    const float* __restrict__ ap,     // [CC, TT]
    const float* __restrict__ bp,     // [CC, TT]
    float* __restrict__ lp_ws)        // [CC, NS]
{
    __shared__ __align__(16) float s_z0[TT * DD];
    __shared__ float s_alpha[TT];
    __shared__ float s_beta[TT];

    const int c = blockIdx.y;
    for (int i = threadIdx.x; i < TT * DD; i += 128)
        s_z0[i] = z0[(size_t)c * TT * DD + i];
    if (threadIdx.x < TT) {
        float a = softplus_f(ap[c * TT + threadIdx.x]);
        s_alpha[threadIdx.x] = a;
        s_beta[threadIdx.x]  = softplus_f(bp[c * TT + threadIdx.x]) - a;
    }
    __syncthreads();

    const int n = blockIdx.x * 128 + threadIdx.x;

    v2f zv[DD / 2];
    {
        const v4f* xr = (const v4f*)(x + (size_t)n * DD);
        #pragma unroll
        for (int i = 0; i < DD / 4; ++i) {
            v4f v = xr[i];
            zv[2*i+0] = (v2f){v.x, v.y};
            zv[2*i+1] = (v2f){v.z, v.w};
        }
    }

    float ld = 0.0f;
    #pragma unroll 1
    for (int t = 0; t < TT; ++t) {
        const float alpha = s_alpha[t];
        const float beta  = s_beta[t];
        const v4f* zr     = (const v4f*)(s_z0 + t * DD);

        v2f diffv[DD / 2];
        v2f r2v = {0.0f, 0.0f};
        #pragma unroll
        for (int i = 0; i < DD / 4; ++i) {
            v4f v = zr[i];                         // ds_load_b128 broadcast
            v2f d0 = zv[2*i+0] - (v2f){v.x, v.y};  // v_pk_add_f32 (neg src1)
            v2f d1 = zv[2*i+1] - (v2f){v.z, v.w};
            diffv[2*i+0] = d0;
            diffv[2*i+1] = d1;
            r2v = __builtin_elementwise_fma(d0, d0, r2v);  // v_pk_fma_f32
            r2v = __builtin_elementwise_fma(d1, d1, r2v);
        }
        const float r2 = r2v.x + r2v.y;
        const float r  = __builtin_amdgcn_sqrtf(r2);           // v_sqrt_f32
        const float h  = __builtin_amdgcn_rcpf(alpha + r);     // v_rcp_f32
        const float bh = beta * h;
        const v2f bhv  = {bh, bh};
        #pragma unroll
        for (int i = 0; i < DD / 2; ++i)
            zv[i] = __builtin_elementwise_fma(bhv, diffv[i], zv[i]);  // v_pk_fma_f32
        // bh ~ O(0.1): log(1+x) via raw v_log_f32 is safe (no cancellation regime)
        const float lg1 = __logf(1.0f + bh);
        const float lg2 = __logf(1.0f + bh * (1.0f - h * r));
        ld = fmaf((float)(DD - 1), lg1, ld) + lg2;
    }

    v2f r2zv = {0.0f, 0.0f};
    #pragma unroll
    for (int i = 0; i < DD / 2; ++i)
        r2zv = __builtin_elementwise_fma(zv[i], zv[i], r2zv);
    const float r2z = r2zv.x + r2zv.y;

    lp_ws[(size_t)c * NS + n] = -0.5f * ((float)DD * LOG2PI + r2z) + ld;
}

// ---------------------------------------------------------------------------
// Kernel 2: logits = x @ W + b via V_WMMA_F32_16X16X4_F32.
// One wave per 16x16 output tile; K=64 -> 16 chained WMMAs.
// grid = (NS/16, ceil(CC/16)), block = 32 (one wave)
// ---------------------------------------------------------------------------
__global__ __launch_bounds__(32) void logits_wmma_kernel(
    const float* __restrict__ x,      // [NS, DD] row-major
    const float* __restrict__ W,      // [DD, CC] row-major
    const float* __restrict__ b,      // [CC]
    float* __restrict__ logits)       // [NS, CC]
{
    const int m0   = blockIdx.x * 16;
    const int n0   = blockIdx.y * 16;
    const int lane = threadIdx.x;      // 0..31 (wave32)
    const int half = lane >> 4;        // 0: K pair {0,1}; 1: K pair {2,3}
    const int lm   = lane & 15;

    const int col  = n0 + lm;
    const int colc = (col < CC) ? col : (CC - 1);   // clamp OOB columns (not stored)

    v8f acc = {};
    #pragma unroll
    for (int k = 0; k < DD; k += 4) {
        const int ka = k + half * 2;
        // A frag: A[M=lm][K=ka], A[M=lm][K=ka+1]  (contiguous, 8B aligned)
        v2f a = *(const v2f*)(x + (size_t)(m0 + lm) * DD + ka);
        // B frag: B[K=ka][N=col], B[K=ka+1][N=col]
        v2f bb;
        bb.x = W[(size_t)ka       * CC + colc];
        bb.y = W[(size_t)(ka + 1) * CC + colc];
        acc = __builtin_amdgcn_wmma_f32_16x16x4_f32(
            /*neg_a=*/false, a, /*neg_b=*/false, bb,
            /*c_mod=*/(short)0, acc, /*reuse_a=*/false, /*reuse_b=*/false);
    }

    if (col < CC) {
        const float bias = b[col];
        #pragma unroll
        for (int i = 0; i < 8; ++i) {
            const int row = m0 + i + half * 8;   // C/D layout: VGPR i -> M=i (+8 for upper lanes)
            logits[(size_t)row * CC + col] = acc[i] + bias;
        }
    }
}

// ---------------------------------------------------------------------------
// Kernel 3: finalize. One thread per sample.
//   marg  = logsumexp_c(lp[c,n] + log freq[c])
//   log_ev = min(marg + EV_BUDGET, 10)
//   lse    = logsumexp_c(logits[n,c])
//   out[n, c] = log1p(exp(log_ev + logit - lse));  out[n, C] = lp[label[n], n]
// ---------------------------------------------------------------------------
__global__ __launch_bounds__(256) void finalize_kernel(
    const float* __restrict__ lp,      // [CC, NS]
    const float* __restrict__ logits,  // [NS, CC]
    const int*   __restrict__ labels,  // [NS]
    const float* __restrict__ freq,    // [CC]
    float* __restrict__ out)           // [NS, CC+1]
{
    __shared__ float s_lfreq[CC];
    if (threadIdx.x < CC)
        s_lfreq[threadIdx.x] = logf(freq[threadIdx.x]);
    __syncthreads();

    const int n = blockIdx.x * 256 + threadIdx.x;
    if (n >= NS) return;

    // logsumexp over classes of lp + log freq (coalesced: stride NS across c)
    float m = -INFINITY;
    for (int c = 0; c < CC; ++c)
        m = fmaxf(m, lp[(size_t)c * NS + n] + s_lfreq[c]);
    float s = 0.0f;
    for (int c = 0; c < CC; ++c)
        s += expf(lp[(size_t)c * NS + n] + s_lfreq[c] - m);
    const float marg   = m + logf(s);
    const float log_ev = fminf(marg + EV_BUDGET_F, LOG_EV_CLAMP_F);

    const float class_lp = lp[(size_t)labels[n] * NS + n];

    // logsumexp over logits row (L2-resident; re-read passes are cheap)
    const float* lrow = logits + (size_t)n * CC;
    float lm = -INFINITY;
    for (int c = 0; c < CC; ++c)
        lm = fmaxf(lm, lrow[c]);
    float ls = 0.0f;
    for (int c = 0; c < CC; ++c)
        ls += expf(lrow[c] - lm);
    const float lse = lm + logf(ls);

    float* orow = out + (size_t)n * (CC + 1);
    for (int c = 0; c < CC; ++c)
        orow[c] = log1pf(expf(log_ev + lrow[c] - lse));   // log(1 + ev*prob_c)
    orow[CC] = class_lp;
}

// ---------------------------------------------------------------------------
extern "C" void kernel_launch(void* const* d_in, const int* in_sizes, int n_in,
                              void* d_out, int out_size, void* d_ws, size_t ws_size,
                              hipStream_t stream) {
    const float* x    = (const float*)d_in[0];   // [NS, DD]
    const int*   lab  = (const int*)  d_in[1];   // [NS]
    const float* freq = (const float*)d_in[2];   // [CC]
    const float* z0   = (const float*)d_in[3];   // [CC, TT, DD]
    const float* ap   = (const float*)d_in[4];   // [CC, TT]
    const float* bp   = (const float*)d_in[5];   // [CC, TT]
    const float* W    = (const float*)d_in[6];   // [DD, CC]
    const float* b    = (const float*)d_in[7];   // [CC]
    float* out = (float*)d_out;                  // [NS, CC+1]

    float* lp_ws     = (float*)d_ws;             // CC*NS floats (3.3 MB)
    float* logits_ws = lp_ws + (size_t)CC * NS;  // NS*CC floats (3.3 MB)

    // 1) flow log-probs: (NS/128) x CC blocks, 128 threads (4 waves, 1 class/wave)
    flow_lp_kernel<<<dim3(NS / 128, CC), 128, 0, stream>>>(x, z0, ap, bp, lp_ws);

    // 2) logits GEMM via f32 WMMA: one wave per 16x16 tile
    logits_wmma_kernel<<<dim3(NS / 16, (CC + 15) / 16), 32, 0, stream>>>(x, W, b, logits_ws);

    // 3) reductions + posterior assembly
    finalize_kernel<<<dim3((NS + 255) / 256), 256, 0, stream>>>(lp_ws, logits_ws, lab, freq, out);
}